// YoloLoss_19078244729063
// MI455X (gfx1250) — compile-verified
//
#include <hip/hip_runtime.h>

typedef __attribute__((ext_vector_type(2))) float v2f;
typedef __attribute__((ext_vector_type(8))) float v8f;
typedef __attribute__((ext_vector_type(4))) unsigned int v4u;
typedef __attribute__((ext_vector_type(8))) int v8i;
typedef __attribute__((ext_vector_type(4))) int v4i;

#define THREADS 256
#define CELLS_PER_BLOCK 256
#define CCH 30
#define NCELLS (2048 * 14 * 14)                  // 401408
#define NBLOCKS (NCELLS / CELLS_PER_BLOCK)       // 1568
#define CHUNK_FLOATS (CELLS_PER_BLOCK * CCH)     // 7680
#define CHUNK_VEC4 (CHUNK_FLOATS / 4)            // 1920

#if __has_builtin(__builtin_amdgcn_tensor_load_to_lds)
#define USE_TDM 1
#else
#define USE_TDM 0
#endif

#if USE_TDM
// Issue one TDM descriptor: copy CHUNK_FLOATS contiguous f32 from global -> LDS.
// D# layout per CDNA5 ISA ch.8 (group0: count/lds/global/type, group1: dims).
// This toolchain exposes the 6-arg builtin form:
//   (uint32x4 g0, int32x8 g1, int32x4, int32x4, int32x8, i32 cpol)
__device__ __forceinline__ void tdm_load_chunk(const float* gsrc, unsigned lds_byte_off) {
  const unsigned long long ga = (unsigned long long)(size_t)gsrc;
  v4u g0;
  g0[0] = 1u;                                            // count=1, user mode, no gather
  g0[1] = lds_byte_off;                                  // lds_addr (bytes)
  g0[2] = (unsigned)(ga & 0xFFFFFFFFu);                  // global_addr[31:0]
  g0[3] = (unsigned)((ga >> 32) & 0x01FFFFFFu)           // global_addr[56:32]
          | 0x80000000u;                                 // type=2 ("image") in [127:126]
  v8i g1;
  g1[0] = (int)(2u << 16);                               // wg_mask=0, data_size=2 (4B), no flags
  g1[1] = (int)((CHUNK_FLOATS & 0xFFFFu) << 16);         // [63:48] tensor_dim0 lo16
  g1[2] = (int)(((CHUNK_FLOATS >> 16) & 0xFFFFu)         // [79:64] tensor_dim0 hi16
          | (1u << 16));                                 // [95:80] tensor_dim1 lo16 = 1
  g1[3] = (int)((CHUNK_FLOATS & 0xFFFFu) << 16);         // [127:112] tile_dim0
  g1[4] = 1;                                             // [143:128] tile_dim1=1, tile_dim2=0
  g1[5] = (int)CHUNK_FLOATS;                             // [191:160] tensor_dim0_stride lo32
  g1[6] = 0;                                             // stride hi16, dim1_stride lo16
  g1[7] = 0;
  const v4i z4 = {0, 0, 0, 0};                           // 2-D tensor: groups 2/3 unused
  const v8i z8 = {0, 0, 0, 0, 0, 0, 0, 0};
  __builtin_amdgcn_tensor_load_to_lds(g0, g1, z4, z4, z8, 0);
}
#endif

__global__ __launch_bounds__(THREADS) void yolo_partial(
    const float* __restrict__ pred, const float* __restrict__ targ,
    float* __restrict__ ws)
{
  __shared__ float4 ldsP4[CHUNK_VEC4];
  __shared__ float4 ldsT4[CHUNK_VEC4];
  __shared__ float red[THREADS / 32];

  const int tid = threadIdx.x;
  const long base = (long)blockIdx.x * CHUNK_FLOATS;

#if USE_TDM
  // Wave 0 alone issues the two DMA descriptors (TDM ignores EXEC: must be
  // reached by exactly one wave), waits TENSORcnt, then the barrier
  // publishes LDS to all 8 waves.
  if (tid < 32) {
    const unsigned offP = (unsigned)(size_t)
        (__attribute__((address_space(3))) void*)&ldsP4[0];
    const unsigned offT = (unsigned)(size_t)
        (__attribute__((address_space(3))) void*)&ldsT4[0];
    tdm_load_chunk(pred + base, offP);
    tdm_load_chunk(targ + base, offT);
    __builtin_amdgcn_s_wait_tensorcnt((short)0);
  }
#else
  const float4* gp = (const float4*)(pred + base);
  const float4* gt = (const float4*)(targ + base);
  for (int i = tid; i < CHUNK_VEC4; i += THREADS) ldsP4[i] = gp[i];
  for (int i = tid; i < CHUNK_VEC4; i += THREADS) ldsT4[i] = gt[i];
#endif
  __syncthreads();

  const float* p = ((const float*)ldsP4) + tid * CCH;
  const float* t = ((const float*)ldsT4) + tid * CCH;

  // ---- per-cell YOLO loss (branch-free; EXEC stays all-ones) ----
  const float coo = (t[4] > 0.0f) ? 1.0f : 0.0f;

  // target box 0 -> xyxy
  const float tcx = t[0] / 14.0f, tcy = t[1] / 14.0f;
  const float tx1 = tcx - 0.5f * t[2], ty1 = tcy - 0.5f * t[3];
  const float tx2 = tcx + 0.5f * t[2], ty2 = tcy + 0.5f * t[3];
  const float ta  = (tx2 - tx1) * (ty2 - ty1);

  float iou0, iou1;
  {
    const float cx = p[0] / 14.0f, cy = p[1] / 14.0f;
    const float x1 = cx - 0.5f * p[2], y1 = cy - 0.5f * p[3];
    const float x2 = cx + 0.5f * p[2], y2 = cy + 0.5f * p[3];
    const float a1 = (x2 - x1) * (y2 - y1);
    const float w = fmaxf(fminf(x2, tx2) - fmaxf(x1, tx1), 0.0f);
    const float h = fmaxf(fminf(y2, ty2) - fmaxf(y1, ty1), 0.0f);
    const float inter = w * h;
    iou0 = inter / (a1 + ta - inter);
  }
  {
    const float cx = p[5] / 14.0f, cy = p[6] / 14.0f;
    const float x1 = cx - 0.5f * p[7], y1 = cy - 0.5f * p[8];
    const float x2 = cx + 0.5f * p[7], y2 = cy + 0.5f * p[8];
    const float a1 = (x2 - x1) * (y2 - y1);
    const float w = fmaxf(fminf(x2, tx2) - fmaxf(x1, tx1), 0.0f);
    const float h = fmaxf(fminf(y2, ty2) - fmaxf(y1, ty1), 0.0f);
    const float inter = w * h;
    iou1 = inter / (a1 + ta - inter);
  }

  const bool r1 = iou1 > iou0;   // argmax (first index wins ties)
  const float rp0 = r1 ? p[5] : p[0];
  const float rp1 = r1 ? p[6] : p[1];
  const float rp2 = r1 ? p[7] : p[2];
  const float rp3 = r1 ? p[8] : p[3];
  const float rp4 = r1 ? p[9] : p[4];
  const float np4 = r1 ? p[4] : p[9];
  const float rt0 = r1 ? t[5] : t[0];
  const float rt1 = r1 ? t[6] : t[1];
  const float rt2 = r1 ? t[7] : t[2];
  const float rt3 = r1 ? t[8] : t[3];

  const float dx = rp0 - rt0, dy = rp1 - rt1;
  const float dw = sqrtf(rp2) - sqrtf(rt2);
  const float dh = sqrtf(rp3) - sqrtf(rt3);
  const float loc = dx * dx + dy * dy + dw * dw + dh * dh;

  const float contain     = rp4 * rp4;
  const float not_contain = np4 * np4;

  const float d4 = p[4] - t[4], d9 = p[9] - t[9];
  const float noobj = d4 * d4 + d9 * d9;

  float cls = 0.0f;
  #pragma unroll
  for (int c = 10; c < 30; ++c) { const float d = p[c] - t[c]; cls += d * d; }

  const float cell =
      coo * (5.0f * loc + 2.0f * contain + not_contain + cls) +
      0.5f * (1.0f - coo) * noobj;

  // ---- wave32 reduction via V_WMMA_F32_16X16X4_F32 ----
  // A: lane's loss in K=0/K=2 slot (vgpr0), vgpr1 = 0.  B = ones.
  // D[m][n] = cell[m] + cell[m+16]  for all n.
  v2f a; a[0] = cell; a[1] = 0.0f;
  v2f b; b[0] = 1.0f; b[1] = 1.0f;
  v8f acc = {};
  acc = __builtin_amdgcn_wmma_f32_16x16x4_f32(
      false, a, false, b, (short)0, acc, false, false);

  float s = acc[0] + acc[1] + acc[2] + acc[3] +
            acc[4] + acc[5] + acc[6] + acc[7];
  s += __shfl_xor(s, 16, 32);   // all 32 lanes now hold the wave total

  if ((tid & 31) == 0) red[tid >> 5] = s;
  __syncthreads();
  if (tid == 0) {
    float tot = 0.0f;
    #pragma unroll
    for (int i = 0; i < THREADS / 32; ++i) tot += red[i];
    ws[blockIdx.x] = tot;
  }
}

__global__ __launch_bounds__(256) void yolo_final(
    const float* __restrict__ ws, float* __restrict__ out)
{
  __shared__ float buf[256];
  const int tid = threadIdx.x;
  float s = 0.0f;
  for (int i = tid; i < NBLOCKS; i += 256) s += ws[i];  // fixed order: deterministic
  buf[tid] = s;
  __syncthreads();
  for (int k = 128; k > 0; k >>= 1) {
    if (tid < k) buf[tid] += buf[tid + k];
    __syncthreads();
  }
  if (tid == 0) out[0] = buf[0] * (1.0f / 2048.0f);
}

extern "C" void kernel_launch(void* const* d_in, const int* in_sizes, int n_in,
                              void* d_out, int out_size, void* d_ws, size_t ws_size,
                              hipStream_t stream) {
  const float* pred = (const float*)d_in[0];
  const float* targ = (const float*)d_in[1];
  float* ws  = (float*)d_ws;
  float* out = (float*)d_out;

  yolo_partial<<<NBLOCKS, THREADS, 0, stream>>>(pred, targ, ws);
  yolo_final<<<1, 256, 0, stream>>>(ws, out);
}